// QKRoPEKVCacheTestModel_66039417143606
// MI455X (gfx1250) — compile-verified
//
#include <hip/hip_runtime.h>

// Problem constants (match reference)
#define T_TOKENS   16384
#define NUM_HEADS  32
#define NUM_KV     8
#define HEAD_SIZE  128
#define NUM_SLOTS  32768
#define Q_ROW      (NUM_HEADS * HEAD_SIZE)   // 4096 floats / token
#define KV_ROW     (NUM_KV * HEAD_SIZE)      // 1024 floats / token or slot
#define LN_BASE    9.2103403719761836f       // ln(10000)

typedef float v4f __attribute__((ext_vector_type(4)));

// gfx1250 async global<->LDS path (ASYNCcnt). The builtins take pointers to
// 16-byte int vectors in explicit address spaces:
//   (int4 AS(1)* gaddr, int4 AS(3)* lds, imm offset, imm cpol)
#if defined(__gfx1250__) && \
    __has_builtin(__builtin_amdgcn_global_load_async_to_lds_b128) && \
    __has_builtin(__builtin_amdgcn_global_store_async_from_lds_b128) && \
    __has_builtin(__builtin_amdgcn_s_wait_asynccnt)
#define USE_ASYNC_LDS 1
typedef int v4i_b128 __attribute__((vector_size(16)));
typedef __attribute__((address_space(1))) v4i_b128* p_g_b128;
typedef __attribute__((address_space(3))) v4i_b128* p_l_b128;
#else
#define USE_ASYNC_LDS 0
#endif

// ---------------------------------------------------------------------------
// Fused kernel (emitted first so the disasm snippet shows it):
//   blocks [0, T)             : RoPE q/k, write q_rot/k_rot/v outputs, scatter
//                               k_rot/v into the cache outputs (4KB coalesced
//                               rows). V is staged via async global->LDS and
//                               drained via async LDS->global (ASYNCcnt).
//   blocks [T, T+NUM_SLOTS)   : copy cache slot (k & v) from input cache to
//                               output cache ONLY if not overwritten (flag==0).
// ---------------------------------------------------------------------------
__global__ __launch_bounds__(256) void rope_kvcache_fused_kernel(
    const float* __restrict__ q, const float* __restrict__ k,
    const float* __restrict__ v,
    const int* __restrict__ positions, const int* __restrict__ slot_mapping,
    const float* __restrict__ kc_in, const float* __restrict__ vc_in,
    float* __restrict__ q_out, float* __restrict__ k_out,
    float* __restrict__ v_out, float* __restrict__ kc_out,
    float* __restrict__ vc_out,
    const int* __restrict__ flags) {

    const int tid = threadIdx.x;

    // ---------------- background cache-slot copy path ----------------
    if (blockIdx.x >= T_TOKENS) {
        const int slot = blockIdx.x - T_TOKENS;
        if (flags[slot]) return;                    // token block owns this slot
        const size_t off = (size_t)slot * KV_ROW;
        v4f dk = __builtin_nontemporal_load((const v4f*)(kc_in + off) + tid);
        v4f dv = __builtin_nontemporal_load((const v4f*)(vc_in + off) + tid);
        __builtin_nontemporal_store(dk, (v4f*)(kc_out + off) + tid);
        __builtin_nontemporal_store(dv, (v4f*)(vc_out + off) + tid);
        return;
    }

    // ---------------------- per-token RoPE path ----------------------
    __shared__ __align__(16) float s_cos[HEAD_SIZE / 2];
    __shared__ __align__(16) float s_sin[HEAD_SIZE / 2];
    __shared__ __align__(16) float s_v[KV_ROW];

    const int token = blockIdx.x;
    const int pos   = positions[token];
    const int slot  = slot_mapping[token];

    const float* vt = v + (size_t)token * KV_ROW;

    // Stage V tile (4KB) into LDS: one async b128 per lane, overlapped with
    // the RoPE math below (tracked with ASYNCcnt on gfx1250).
#if USE_ASYNC_LDS
    __builtin_amdgcn_global_load_async_to_lds_b128(
        (p_g_b128)(vt + tid * 4), (p_l_b128)(s_v + tid * 4), 0, 0);
#else
    *(v4f*)(s_v + tid * 4) = *(const v4f*)(vt + tid * 4);
#endif

    // cos/sin table: one entry per rotary pair, shared by all heads & waves.
    if (tid < HEAD_SIZE / 2) {
        float inv_freq = __expf(-(float)tid * (LN_BASE / 64.0f));
        float f = (float)pos * inv_freq;
        float sn, cs;
        __sincosf(f, &sn, &cs);
        s_cos[tid] = cs;
        s_sin[tid] = sn;
    }
    __syncthreads();

    // ---- Q RoPE: 32 heads * 64 pairs = 2048 pairs; 4 pairs/lane, 2 iters ----
    const float* qt = q + (size_t)token * Q_ROW;
    float*       qo = q_out + (size_t)token * Q_ROW;
#pragma unroll
    for (int it = 0; it < 2; ++it) {
        int p = it * 1024 + tid * 4;        // pair index (stays within a head)
        int h = p >> 6;
        int i = p & 63;
        const float* base = qt + h * HEAD_SIZE + i;
        v4f x1 = __builtin_nontemporal_load((const v4f*)base);
        v4f x2 = __builtin_nontemporal_load((const v4f*)(base + 64));
        v4f c  = *(const v4f*)(s_cos + i);
        v4f s  = *(const v4f*)(s_sin + i);
        v4f o1 = x1 * c - x2 * s;
        v4f o2 = x2 * c + x1 * s;
        __builtin_nontemporal_store(o1, (v4f*)(qo + h * HEAD_SIZE + i));
        __builtin_nontemporal_store(o2, (v4f*)(qo + h * HEAD_SIZE + 64 + i));
    }

    // ---- K RoPE: 8 heads * 64 pairs = 512 pairs; lanes 0..127 ----
    const size_t slot_off = (size_t)slot * KV_ROW;
    if (tid < 128) {
        int p = tid * 4;
        int h = p >> 6;
        int i = p & 63;
        const float* kt = k + (size_t)token * KV_ROW + h * HEAD_SIZE + i;
        v4f x1 = __builtin_nontemporal_load((const v4f*)kt);
        v4f x2 = __builtin_nontemporal_load((const v4f*)(kt + 64));
        v4f c  = *(const v4f*)(s_cos + i);
        v4f s  = *(const v4f*)(s_sin + i);
        v4f o1 = x1 * c - x2 * s;
        v4f o2 = x2 * c + x1 * s;
        float* ko = k_out + (size_t)token * KV_ROW + h * HEAD_SIZE + i;
        float* kc = kc_out + slot_off + h * HEAD_SIZE + i;
        __builtin_nontemporal_store(o1, (v4f*)ko);
        __builtin_nontemporal_store(o2, (v4f*)(ko + 64));
        __builtin_nontemporal_store(o1, (v4f*)kc);
        __builtin_nontemporal_store(o2, (v4f*)(kc + 64));
    }

    // ---- V pass-through + scatter: drain the LDS stage ----
#if USE_ASYNC_LDS
    __builtin_amdgcn_s_wait_asynccnt(0);    // our wave's async loads landed
#endif
    __syncthreads();                        // all waves' LDS writes visible

#if USE_ASYNC_LDS
    __builtin_amdgcn_global_store_async_from_lds_b128(
        (p_g_b128)(v_out + (size_t)token * KV_ROW + tid * 4),
        (p_l_b128)(s_v + tid * 4), 0, 0);
    __builtin_amdgcn_global_store_async_from_lds_b128(
        (p_g_b128)(vc_out + slot_off + tid * 4),
        (p_l_b128)(s_v + tid * 4), 0, 0);
    // implicit S_WAIT_IDLE at S_ENDPGM covers outstanding ASYNCcnt
#else
    v4f vv = *(const v4f*)(s_v + tid * 4);
    __builtin_nontemporal_store(vv, (v4f*)(v_out + (size_t)token * KV_ROW + tid * 4));
    __builtin_nontemporal_store(vv, (v4f*)(vc_out + slot_off + tid * 4));
#endif
}

// ---------------------------------------------------------------------------
// Kernel: mark which cache slots get overwritten by this batch of tokens.
// ---------------------------------------------------------------------------
__global__ __launch_bounds__(256) void mark_slots_kernel(
    const int* __restrict__ slot_mapping, int* __restrict__ flags) {
    int t = blockIdx.x * 256 + threadIdx.x;   // grid sized exactly to T_TOKENS
    flags[slot_mapping[t]] = 1;
}

// ---------------------------------------------------------------------------
// Launch
// ---------------------------------------------------------------------------
extern "C" void kernel_launch(void* const* d_in, const int* in_sizes, int n_in,
                              void* d_out, int out_size, void* d_ws, size_t ws_size,
                              hipStream_t stream) {
    const float* q            = (const float*)d_in[0];
    const float* k            = (const float*)d_in[1];
    const float* v            = (const float*)d_in[2];
    const int*   positions    = (const int*)d_in[3];
    const int*   slot_mapping = (const int*)d_in[4];
    const float* kc_in        = (const float*)d_in[5];
    const float* vc_in        = (const float*)d_in[6];

    float* out    = (float*)d_out;
    float* q_out  = out;
    float* k_out  = q_out + (size_t)T_TOKENS * Q_ROW;
    float* v_out  = k_out + (size_t)T_TOKENS * KV_ROW;
    float* kc_out = v_out + (size_t)T_TOKENS * KV_ROW;
    float* vc_out = kc_out + (size_t)NUM_SLOTS * KV_ROW;

    if (ws_size >= (size_t)NUM_SLOTS * sizeof(int)) {
        // Fast path: occupancy bitmap -> write every cache byte exactly once.
        int* flags = (int*)d_ws;
        (void)hipMemsetAsync(flags, 0, (size_t)NUM_SLOTS * sizeof(int), stream);
        mark_slots_kernel<<<T_TOKENS / 256, 256, 0, stream>>>(slot_mapping, flags);
        rope_kvcache_fused_kernel<<<T_TOKENS + NUM_SLOTS, 256, 0, stream>>>(
            q, k, v, positions, slot_mapping, kc_in, vc_in,
            q_out, k_out, v_out, kc_out, vc_out, flags);
    } else {
        // Fallback: bulk-copy caches, then token blocks only (slot-copy blocks
        // never launched, so `flags` is never dereferenced).
        (void)hipMemcpyAsync(kc_out, kc_in,
                             (size_t)NUM_SLOTS * KV_ROW * sizeof(float),
                             hipMemcpyDeviceToDevice, stream);
        (void)hipMemcpyAsync(vc_out, vc_in,
                             (size_t)NUM_SLOTS * KV_ROW * sizeof(float),
                             hipMemcpyDeviceToDevice, stream);
        rope_kvcache_fused_kernel<<<T_TOKENS, 256, 0, stream>>>(
            q, k, v, positions, slot_mapping, kc_in, vc_in,
            q_out, k_out, v_out, kc_out, vc_out, (const int*)d_ws);
    }
}